// GNNMultiEdgeset_25340307046681
// MI455X (gfx1250) — compile-verified
//
#include <hip/hip_runtime.h>
#include <math.h>

#define HDIM    128
#define LAYERS  3
#define OUTD    64
#define NGRAPHS 64

typedef __attribute__((ext_vector_type(2))) float v2f;
typedef __attribute__((ext_vector_type(8))) float v8f;

__device__ __forceinline__ float gelu_exact(float x) {
    return 0.5f * x * (1.0f + erff(x * 0.70710678118654752440f));
}

__device__ __forceinline__ void atomAddF(float* p, float v) {
    unsafeAtomicAdd(p, v);   // global_atomic_add_f32
}

// ---------------------------------------------------------------------------
// Generic f32 WMMA GEMM: C = act(A[M,K] @ W[K,Nc] + bias)
// One wave -> one 16x16 tile of C. Uses V_WMMA_F32_16X16X4_F32.
// A 16x4 layout: lanes 0-15 rows M, VGPR0/1 = K{0,1}; lanes 16-31 = K{2,3}.
// B 4x16 layout: lanes 0-15 cols N, VGPR0/1 = K{0,1}; lanes 16-31 = K{2,3}.
// C/D: VGPR v, lanes 0-15 -> M=v, lanes 16-31 -> M=v+8; N = lane&15.
// ---------------------------------------------------------------------------
template <int GELU_OUT>
__global__ __launch_bounds__(256) void gemm_wmma_f32(
    const float* __restrict__ A, const float* __restrict__ W,
    const float* __restrict__ bias, float* __restrict__ C,
    int M, int K, int Nc) {
    const int lane   = threadIdx.x & 31;
    const int wave   = threadIdx.x >> 5;
    const int tilesN = Nc >> 4;
    const int tile   = blockIdx.x * (blockDim.x >> 5) + wave;
    const int nTiles = (M >> 4) * tilesN;
    if (tile >= nTiles) return;                 // wave-uniform: EXEC stays all-1s
    const int tm   = (tile / tilesN) << 4;
    const int tn   = (tile % tilesN) << 4;
    const int half = lane >> 4;                 // selects K pair {0,1} vs {2,3}
    const int lid  = lane & 15;

    v8f acc = {};
    const float* aptr = A + (size_t)(tm + lid) * K + (half << 1);
    const float* bptr = W + (size_t)(half << 1) * Nc + tn + lid;
    for (int k = 0; k < K; k += 4) {
        v2f a, b;
        a.x = aptr[0];
        a.y = aptr[1];
        b.x = bptr[0];
        b.y = bptr[Nc];
        acc = __builtin_amdgcn_wmma_f32_16x16x4_f32(false, a, false, b,
                                                    (short)0, acc, false, false);
        aptr += 4;
        bptr += (size_t)4 * Nc;
    }
    const int   col  = tn + lid;
    const float bv   = bias[col];
    const int   row0 = tm + (half << 3);
#pragma unroll
    for (int v = 0; v < 8; ++v) {
        float val = acc[v] + bv;
        if (GELU_OUT) val = gelu_exact(val);
        C[(size_t)(row0 + v) * Nc + col] = val;
    }
}

// ---------------------------------------------------------------------------
__global__ void zero_f(float* __restrict__ p, size_t n) {
    size_t i = (size_t)blockIdx.x * blockDim.x + threadIdx.x;
    if (i < n) p[i] = 0.0f;
}

__global__ void broadcast_x(const float* __restrict__ x, float* __restrict__ h,
                            size_t nx, size_t total) {
    size_t i = (size_t)blockIdx.x * blockDim.x + threadIdx.x;
    if (i < total) h[i] = x[i % nx];
}

__global__ void edge_mask_kernel(const float* __restrict__ nmask,  // [4,N]
                                 const int* __restrict__ src,
                                 const int* __restrict__ dst,
                                 float* __restrict__ emask,        // [4,E]
                                 int E, int N) {
    size_t i = (size_t)blockIdx.x * blockDim.x + threadIdx.x;
    if (i >= (size_t)4 * E) return;
    int rc = (int)(i / E), e = (int)(i % E);
    emask[i] = nmask[(size_t)rc * N + src[e]] * nmask[(size_t)rc * N + dst[e]];
}

// msg = gelu(h[rc,src]+emb)*mask, atomically scattered into agg[rc,dst].
// One wave per (edge, replica); 4 floats per lane (128 = 32*4).
__global__ __launch_bounds__(256) void msg_scatter(
    const float* __restrict__ h, const float* __restrict__ emb,
    const float* __restrict__ emask, const int* __restrict__ src,
    const int* __restrict__ dst, float* __restrict__ agg, int E, int N) {
    int wid = blockIdx.x * (blockDim.x >> 5) + (threadIdx.x >> 5);
    if (wid >= 4 * E) return;
    int e = wid >> 2, rc = wid & 3;
    int lane = threadIdx.x & 31;
    float m = emask[(size_t)rc * E + e];
    int s = src[e], d = dst[e];
    float4 hx = *((const float4*)(h + ((size_t)rc * N + s) * HDIM) + lane);
    float4 ex = *((const float4*)(emb + (size_t)e * HDIM) + lane);
    float4 r;
    r.x = gelu_exact(hx.x + ex.x) * m;
    r.y = gelu_exact(hx.y + ex.y) * m;
    r.z = gelu_exact(hx.z + ex.z) * m;
    r.w = gelu_exact(hx.w + ex.w) * m;
    float* ap = agg + ((size_t)rc * N + d) * HDIM + lane * 4;
    atomAddF(ap + 0, r.x);
    atomAddF(ap + 1, r.y);
    atomAddF(ap + 2, r.z);
    atomAddF(ap + 3, r.w);
}

__global__ void combine_eps(const float* __restrict__ h,
                            const float* __restrict__ agg,
                            const float* __restrict__ eps, int layer,
                            float* __restrict__ out, size_t total) {
    size_t i = (size_t)blockIdx.x * blockDim.x + threadIdx.x;
    if (i < total) out[i] = (1.0f + eps[layer]) * h[i] + agg[i];
}

// Masked segment-sum: one wave per row (rc*N+n), 2 cols/lane over OUTD=64.
__global__ __launch_bounds__(256) void pool_scatter(
    const float* __restrict__ y, const float* __restrict__ nmask,  // [4,N]
    const int* __restrict__ batch, float* __restrict__ num,        // [256,64]
    float* __restrict__ den, int N) {
    int row = blockIdx.x * (blockDim.x >> 5) + (threadIdx.x >> 5);
    if (row >= 4 * N) return;
    int rc = row / N, n = row % N;
    int lane = threadIdx.x & 31;
    float fm = nmask[row];
    int seg = batch[n] + rc * NGRAPHS;
    float2 yv = *((const float2*)(y + (size_t)row * OUTD) + lane);
    float* np = num + (size_t)seg * OUTD + lane * 2;
    atomAddF(np + 0, yv.x * fm);
    atomAddF(np + 1, yv.y * fm);
    if (lane == 0) atomAddF(den + seg, fm);
}

__global__ void pool_finalize(const float* __restrict__ num,
                              const float* __restrict__ den,
                              float* __restrict__ out, int total) {
    int i = blockIdx.x * blockDim.x + threadIdx.x;
    if (i < total) out[i] = num[i] / fmaxf(den[i >> 6], 1e-12f);
}

// ---------------------------------------------------------------------------
static void launch_gemm(const float* A, const float* W, const float* b, float* C,
                        int M, int K, int Nc, bool geluOut, hipStream_t stream) {
    int tiles  = (M / 16) * (Nc / 16);
    int blocks = (tiles + 7) / 8;
    if (geluOut)
        gemm_wmma_f32<1><<<blocks, 256, 0, stream>>>(A, W, b, C, M, K, Nc);
    else
        gemm_wmma_f32<0><<<blocks, 256, 0, stream>>>(A, W, b, C, M, K, Nc);
}

extern "C" void kernel_launch(void* const* d_in, const int* in_sizes, int n_in,
                              void* d_out, int out_size, void* d_ws, size_t ws_size,
                              hipStream_t stream) {
    const float* x         = (const float*)d_in[0];
    const float* edge_attr = (const float*)d_in[1];
    const float* node_mask = (const float*)d_in[2];   // [4,N]
    const float* bond_W    = (const float*)d_in[3];
    const float* bond_b    = (const float*)d_in[4];
    const float* eps       = (const float*)d_in[5];
    const float* conv_W1   = (const float*)d_in[6];
    const float* conv_b1   = (const float*)d_in[7];
    const float* conv_W2   = (const float*)d_in[8];
    const float* conv_b2   = (const float*)d_in[9];
    const float* mlp_W1    = (const float*)d_in[10];
    const float* mlp_b1    = (const float*)d_in[11];
    const float* mlp_W2    = (const float*)d_in[12];
    const float* mlp_b2    = (const float*)d_in[13];
    const int*   edge_index= (const int*)d_in[14];
    const int*   batch     = (const int*)d_in[15];

    const int N  = in_sizes[0] / HDIM;
    const int E  = in_sizes[1] / HDIM;
    const int M  = 4 * N;                       // R*C*N rows
    const int*   src = edge_index;
    const int*   dst = edge_index + E;

    float* ws = (float*)d_ws;
    size_t off = 0;
    float* emb   = ws + off; off += (size_t)E * HDIM;
    float* h     = ws + off; off += (size_t)M * HDIM;
    float* agg   = ws + off; off += (size_t)M * HDIM;
    float* tA    = ws + off; off += (size_t)M * HDIM;
    float* tB    = ws + off; off += (size_t)M * HDIM;
    float* emask = ws + off; off += (size_t)4 * E;
    float* ybuf  = ws + off; off += (size_t)M * OUTD;
    float* numb  = ws + off; off += (size_t)4 * NGRAPHS * OUTD;
    float* denb  = ws + off; off += (size_t)4 * NGRAPHS;
    (void)ws_size; (void)n_in;

    const size_t hTot = (size_t)M * HDIM;

    // h = broadcast(x) over 4 replicas
    broadcast_x<<<(unsigned)((hTot + 255) / 256), 256, 0, stream>>>(
        x, h, (size_t)N * HDIM, hTot);
    // edge_mask[rc,e] = nmask[rc,src]*nmask[rc,dst]
    edge_mask_kernel<<<(unsigned)(((size_t)4 * E + 255) / 256), 256, 0, stream>>>(
        node_mask, src, dst, emask, E, N);

    for (int l = 0; l < LAYERS; ++l) {
        // emb = edge_attr @ bond_W[l] + bond_b[l]
        launch_gemm(edge_attr, bond_W + (size_t)l * HDIM * HDIM,
                    bond_b + (size_t)l * HDIM, emb, E, HDIM, HDIM, false, stream);
        zero_f<<<(unsigned)((hTot + 255) / 256), 256, 0, stream>>>(agg, hTot);
        // scatter messages
        msg_scatter<<<(unsigned)(((size_t)4 * E + 7) / 8), 256, 0, stream>>>(
            h, emb, emask, src, dst, agg, E, N);
        // out = (1+eps)*h + agg
        combine_eps<<<(unsigned)((hTot + 255) / 256), 256, 0, stream>>>(
            h, agg, eps, l, tA, hTot);
        // h = gelu( gelu(out@W1+b1) @ W2 + b2 )
        launch_gemm(tA, conv_W1 + (size_t)l * HDIM * HDIM,
                    conv_b1 + (size_t)l * HDIM, tB, M, HDIM, HDIM, true, stream);
        launch_gemm(tB, conv_W2 + (size_t)l * HDIM * HDIM,
                    conv_b2 + (size_t)l * HDIM, h, M, HDIM, HDIM, true, stream);
    }

    // MLP head: y = gelu(h@mlp_W1+b1) @ mlp_W2 + b2
    launch_gemm(h, mlp_W1, mlp_b1, tA, M, HDIM, HDIM, true, stream);
    launch_gemm(tA, mlp_W2, mlp_b2, ybuf, M, HDIM, OUTD, false, stream);

    // masked segment mean
    const size_t poolN = (size_t)4 * NGRAPHS * OUTD + (size_t)4 * NGRAPHS;
    zero_f<<<(unsigned)((poolN + 255) / 256), 256, 0, stream>>>(numb, poolN);
    pool_scatter<<<(unsigned)(((size_t)M + 7) / 8), 256, 0, stream>>>(
        ybuf, node_mask, batch, numb, denb, N);
    pool_finalize<<<(unsigned)((4 * NGRAPHS * OUTD + 255) / 256), 256, 0, stream>>>(
        numb, denb, (float*)d_out, 4 * NGRAPHS * OUTD);
    (void)out_size;
}